// S4Convolutional_40578851012834
// MI455X (gfx1250) — compile-verified
//
#include <hip/hip_runtime.h>
#include <hip/hip_bf16.h>

// S4 convolutional model for MI455X (gfx1250, wave32).
// Pipeline: K1 input projection -> K2 expm + log-depth kernel gen ->
//           K3 WMMA (f32 16x16x4) causal convolution (unrolled x4) ->
//           K4 rank-1 expansion.

typedef __attribute__((ext_vector_type(2))) float v2f;
typedef __attribute__((ext_vector_type(8))) float v8f;

#define SEQ    4096
#define NBATCH 16
#define DMODEL 256
#define NSTATE 64
#define NA     65      // augmented [[dA, dtB],[0,0]]
#define NSTR   66      // LDS row stride (floats)

// ---------------------------------------------------------------------------
// K1: up[b,l] = dot(u[b,l,:], W_in).  One wave per (b,l) row, coalesced.
// Stores TRANSPOSED: upT[l*16 + b]  (so WMMA A-fragments load contiguously).
// ---------------------------------------------------------------------------
__global__ __launch_bounds__(256) void k_proj(const float* __restrict__ u,
                                              const float* __restrict__ Win,
                                              float* __restrict__ upT) {
  __shared__ float w[DMODEL];
  const int tid = threadIdx.x;
  w[tid] = Win[tid];
  __syncthreads();
  const int wave = tid >> 5, lane = tid & 31;
  const int row = blockIdx.x * 8 + wave;            // 0..65535 = b*4096 + l
  const float* ur = u + (size_t)row * DMODEL;
  float s = 0.f;
#pragma unroll
  for (int j = 0; j < 8; ++j) {
    const int c = lane + 32 * j;
    s += ur[c] * w[c];
  }
#pragma unroll
  for (int off = 16; off > 0; off >>= 1) s += __shfl_xor(s, off, 32);
  if (lane == 0) {
    const int b = row >> 12, l = row & (SEQ - 1);
    upT[l * NBATCH + b] = s;
  }
}

// ---------------------------------------------------------------------------
// K2: single workgroup (1024 threads).
//  exp([[dA, dt*B],[0,0]]) = [[A_bar, B_bar],[0,1]]   (scaling/squaring s=5,
//  degree-8 Horner Taylor).  Then k[64j+i] = (c^T (A_bar^64)^j) (A_bar^i b_bar)
//  via log-depth doubling -> K = W @ V -> kpad[16..16+4096), front-padded 0s.
// ---------------------------------------------------------------------------
__global__ __launch_bounds__(1024) void k_kernelgen(const float* __restrict__ A,
                                                    const float* __restrict__ Bv,
                                                    const float* __restrict__ Cv,
                                                    const float* __restrict__ logdt,
                                                    float* __restrict__ kpad) {
  extern __shared__ float lds[];
  float* Xs = lds;                  // augmented dA / 32
  float* Ra = lds + 1 * NA * NSTR;  // ping
  float* Rb = lds + 2 * NA * NSTR;  // pong
  float* V  = lds + 3 * NA * NSTR;  // V[:,i] = A_bar^i b_bar
  float* W  = lds + 4 * NA * NSTR;  // W[j,:] = c^T (A_bar^64)^j
  const int tid = threadIdx.x;
  const int nthr = blockDim.x;

  float dt = expf(logdt[0]);
  dt = fminf(fmaxf(dt, 0.001f), 0.1f);

  // Build Xs = [[dt*A, dt*B],[0,0]] / 32
  for (int e = tid; e < NA * NA; e += nthr) {
    const int i = e / NA, j = e % NA;
    float v = 0.f;
    if (i < NSTATE) v = (j < NSTATE) ? dt * A[i * NSTATE + j] : dt * Bv[i];
    Xs[i * NSTR + j] = v * (1.0f / 32.0f);
  }
  __syncthreads();

  // Horner Taylor degree 8: R = I + X(I + X/2(...(I + X/8)...))
  for (int e = tid; e < NA * NA; e += nthr) {
    const int i = e / NA, j = e % NA;
    Ra[i * NSTR + j] = Xs[i * NSTR + j] * 0.125f + (i == j ? 1.f : 0.f);
  }
  __syncthreads();
  float* rc = Ra; float* rn = Rb;
  for (int h = 7; h >= 1; --h) {
    const float inv = 1.0f / (float)h;
    for (int e = tid; e < NA * NA; e += nthr) {
      const int i = e / NA, j = e % NA;
      float s = 0.f;
      for (int kk = 0; kk < NA; ++kk) s += Xs[i * NSTR + kk] * rc[kk * NSTR + j];
      rn[i * NSTR + j] = s * inv + (i == j ? 1.f : 0.f);
    }
    __syncthreads();
    float* t = rc; rc = rn; rn = t;
  }
  // 5 squarings: exp(G) = (exp(G/32))^32
  for (int sq = 0; sq < 5; ++sq) {
    for (int e = tid; e < NA * NA; e += nthr) {
      const int i = e / NA, j = e % NA;
      float s = 0.f;
      for (int kk = 0; kk < NA; ++kk) s += rc[i * NSTR + kk] * rc[kk * NSTR + j];
      rn[i * NSTR + j] = s;
    }
    __syncthreads();
    float* t = rc; rc = rn; rn = t;
  }

  // rc = exp(G): A_bar = rc[0:64,0:64], b_bar = rc[0:64,64]
  if (tid < NSTATE) V[tid * NSTR + 0] = rc[tid * NSTR + NSTATE];
  if (tid >= 64 && tid < 128) W[0 * NSTR + (tid - 64)] = Cv[tid - 64];
  __syncthreads();

  // V doubling: P starts as A_bar (in rc); after 6 iters P = A_bar^64.
  int w = 1;
  for (int p = 0; p < 6; ++p) {
    const int total = NSTATE * w;
    for (int e = tid; e < total; e += nthr) {      // V[:, w+t] = P @ V[:, t]
      const int i = e / w, t2 = e % w;
      float s = 0.f;
      for (int kk = 0; kk < NSTATE; ++kk) s += rc[i * NSTR + kk] * V[kk * NSTR + t2];
      V[i * NSTR + w + t2] = s;
    }
    __syncthreads();
    for (int e = tid; e < NSTATE * NSTATE; e += nthr) {   // P = P @ P
      const int i = e >> 6, j = e & 63;
      float s = 0.f;
      for (int kk = 0; kk < NSTATE; ++kk) s += rc[i * NSTR + kk] * rc[kk * NSTR + j];
      rn[i * NSTR + j] = s;
    }
    __syncthreads();
    float* t = rc; rc = rn; rn = t;
    w <<= 1;
  }

  // W doubling with Q = A_bar^64 (in rc)
  w = 1;
  for (int p = 0; p < 6; ++p) {
    const int total = w * NSTATE;
    for (int e = tid; e < total; e += nthr) {      // W[w+t, :] = W[t, :] @ Q
      const int t2 = e >> 6, j = e & 63;
      float s = 0.f;
      for (int kk = 0; kk < NSTATE; ++kk) s += W[t2 * NSTR + kk] * rc[kk * NSTR + j];
      W[(w + t2) * NSTR + j] = s;
    }
    __syncthreads();
    if (p < 5) {                                   // Q = Q @ Q
      for (int e = tid; e < NSTATE * NSTATE; e += nthr) {
        const int i = e >> 6, j = e & 63;
        float s = 0.f;
        for (int kk = 0; kk < NSTATE; ++kk) s += rc[i * NSTR + kk] * rc[kk * NSTR + j];
        rn[i * NSTR + j] = s;
      }
      __syncthreads();
      float* t = rc; rc = rn; rn = t;
    }
    w <<= 1;
  }

  // k[64j + i] = W[j,:] . V[:,i]  -> kpad (16-float zero front pad)
  for (int e = tid; e < NSTATE * NSTATE; e += nthr) {
    const int j = e >> 6, i = e & 63;
    float s = 0.f;
    for (int kk = 0; kk < NSTATE; ++kk) s += W[j * NSTR + kk] * V[kk * NSTR + i];
    kpad[16 + j * NSTATE + i] = s;
  }
  if (tid < 16) kpad[tid] = 0.f;
}

// ---------------------------------------------------------------------------
// K3: causal conv via V_WMMA_F32_16X16X4_F32.
//  y[b, L0+n] = sum_t up[b,t] * khat[L0+n-t] + D*up[b,L0+n]
//  M=batch(16), N=16 output positions, K=time chunks of 4.
//  One WG per l-tile (256 WGs); K split over 8 waves, unrolled x4 so the
//  loads of 4 chunks issue as one clause ahead of 4 accumulating WMMAs.
// ---------------------------------------------------------------------------
__device__ __forceinline__ void load_frag(const float* __restrict__ upT,
                                          const float* __restrict__ kb,
                                          int t0, int MN, int Kb,
                                          v2f& a, v2f& b) {
  // A[M][K] = up[M, t0+K]   (lanes 0..15 read 16 consecutive floats per half)
  a.x = upT[(t0 + Kb + 0) * NBATCH + MN];
  a.y = upT[(t0 + Kb + 1) * NBATCH + MN];
  // B[K][N] = khat[L0 + N - t0 - K]  (contiguous per half-wave; pad covers <0)
  b.x = kb[MN - t0 - (Kb + 0)];
  b.y = kb[MN - t0 - (Kb + 1)];
}

__device__ __forceinline__ v8f wmma_acc(v2f a, v2f b, v8f c) {
  return __builtin_amdgcn_wmma_f32_16x16x4_f32(
      /*neg_a=*/false, a, /*neg_b=*/false, b,
      /*c_mod=*/(short)0, c, /*reuse_a=*/false, /*reuse_b=*/false);
}

__global__ __launch_bounds__(256) void k_conv(const float* __restrict__ upT,
                                              const float* __restrict__ kpad,
                                              const float* __restrict__ Dp,
                                              float* __restrict__ y) {
  const int tile = blockIdx.x;            // 0..255
  const int L0 = tile * 16;
  const int tid = threadIdx.x;
  const int wave = tid >> 5, lane = tid & 31;
  const int nchunk = 4 * (tile + 1);      // K-chunks covering t in [0, L0+16)
  const int MN = lane & 15;               // A: row M (batch); B: col N (pos)
  const int Kb = (lane >> 4) << 1;        // K base per half-wave (0 or 2)
  v8f acc = {0.f, 0.f, 0.f, 0.f, 0.f, 0.f, 0.f, 0.f};
  const float* kb = kpad + 16 + L0;       // khat[L0 + .] with front padding

  // All loop conditions are wave-uniform -> EXEC is all-1s at every WMMA.
  int c = wave;
  for (; c + 24 < nchunk; c += 32) {      // 4 chunks per wave per iteration
    v2f a0, b0, a1, b1, a2, b2, a3, b3;
    load_frag(upT, kb, 4 * (c +  0), MN, Kb, a0, b0);
    load_frag(upT, kb, 4 * (c +  8), MN, Kb, a1, b1);
    load_frag(upT, kb, 4 * (c + 16), MN, Kb, a2, b2);
    load_frag(upT, kb, 4 * (c + 24), MN, Kb, a3, b3);
    acc = wmma_acc(a0, b0, acc);
    acc = wmma_acc(a1, b1, acc);
    acc = wmma_acc(a2, b2, acc);
    acc = wmma_acc(a3, b3, acc);
  }
  for (; c < nchunk; c += 8) {            // remainder (0..3 chunks per wave)
    v2f a, b;
    load_frag(upT, kb, 4 * c, MN, Kb, a, b);
    acc = wmma_acc(a, b, acc);
  }

  __shared__ float red[8 * 256];
#pragma unroll
  for (int r = 0; r < 8; ++r) red[wave * 256 + r * 32 + lane] = acc[r];
  __syncthreads();

  if (wave == 0) {
    const float D0 = Dp[0];
#pragma unroll
    for (int r = 0; r < 8; ++r) {
      float s = 0.f;
#pragma unroll
      for (int w2 = 0; w2 < 8; ++w2) s += red[w2 * 256 + r * 32 + lane];
      const int Mr = r + ((lane >> 4) << 3);  // batch (C-matrix layout)
      const int Nr = lane & 15;               // position offset
      s += D0 * upT[(L0 + Nr) * NBATCH + Mr];
      y[Mr * SEQ + L0 + Nr] = s;
    }
  }
}

// ---------------------------------------------------------------------------
// K4: out[b,l,d] = y[b,l] * W_out[d].  Pure-bandwidth float4 stores (64 MiB).
// ---------------------------------------------------------------------------
__global__ __launch_bounds__(256) void k_expand(const float* __restrict__ y,
                                                const float* __restrict__ Wout,
                                                float* __restrict__ out) {
  __shared__ float4 w4[DMODEL / 4];
  const int tid = threadIdx.x;
  if (tid < DMODEL / 4) w4[tid] = ((const float4*)Wout)[tid];
  __syncthreads();
  const size_t idx = (size_t)blockIdx.x * 256 + tid;   // 0 .. 4M-1 (float4)
  const size_t row = idx >> 6;                         // (b,l) flat index
  const int c = (int)(idx & 63);
  const float v = y[row];
  const float4 w = w4[c];
  float4 o;
  o.x = v * w.x; o.y = v * w.y; o.z = v * w.z; o.w = v * w.w;
  ((float4*)out)[idx] = o;
}

// ---------------------------------------------------------------------------
extern "C" void kernel_launch(void* const* d_in, const int* in_sizes, int n_in,
                              void* d_out, int out_size, void* d_ws, size_t ws_size,
                              hipStream_t stream) {
  const float* u     = (const float*)d_in[0];
  const float* A     = (const float*)d_in[1];
  const float* B     = (const float*)d_in[2];
  const float* C     = (const float*)d_in[3];
  const float* D     = (const float*)d_in[4];
  const float* logdt = (const float*)d_in[5];
  const float* Win   = (const float*)d_in[6];
  const float* Wout  = (const float*)d_in[7];
  float* ws   = (float*)d_ws;
  float* kpad = ws;                         // 16 + 4096 floats
  float* upT  = ws + 4112;                  // 4096*16 floats (transposed up)
  float* yb   = ws + 4112 + SEQ * NBATCH;   // 16*4096 floats
  float* out  = (float*)d_out;

  k_proj<<<(NBATCH * SEQ) / 8, 256, 0, stream>>>(u, Win, upT);
  k_kernelgen<<<1, 1024, 5 * NA * NSTR * sizeof(float), stream>>>(A, B, C, logdt, kpad);
  k_conv<<<SEQ / 16, 256, 0, stream>>>(upT, kpad, D, yb);
  k_expand<<<(NBATCH * SEQ * DMODEL) / 4 / 256, 256, 0, stream>>>(yb, Wout, out);
}